// FedGSL_gpr_69320772157905
// MI455X (gfx1250) — compile-verified
//
#include <hip/hip_runtime.h>
#include <hip/hip_bf16.h>

// ---------------------------------------------------------------------------
// FedGSL / GPR-GNN forward for MI455X (gfx1250, wave32, WMMA f32 16x16x4)
//   x0     = relu(X @ W1 + b1) @ W2 + b2                  (WMMA, async-LDS stage)
//   dinv   = rsqrt(deg + selfloop)                        (atomics)
//   hidden = sum_k temp[k] * A_hat^k x0                   (wave-per-edge scatter)
//   out    = concat(hidden, glob) @ Wl + bl               (WMMA)
// ---------------------------------------------------------------------------

#define IN_C 512
#define HID  64
#define LOC  64
#define GLOB 64
#define OUT_C 40
#define KPOW 10

typedef float v2f __attribute__((ext_vector_type(2)));
typedef float v8f __attribute__((ext_vector_type(8)));
typedef int   v4i __attribute__((vector_size(16)));   // matches builtin param pointee

// ---- gfx1250 async global->LDS staging (ASYNCcnt path), with fallback ------
#if __has_builtin(__builtin_amdgcn_global_load_async_to_lds_b128)
#define HAVE_ASYNC_LDS 1
typedef __attribute__((address_space(1))) v4i gvec_t;   // global 16B vector
typedef __attribute__((address_space(3))) v4i lvec_t;   // LDS 16B vector
__device__ __forceinline__ void async_copy_b128(void* lds_dst, const void* gsrc) {
    __builtin_amdgcn_global_load_async_to_lds_b128(
        (gvec_t*)gsrc, (lvec_t*)lds_dst, /*offset=*/0, /*cpol=*/0);
}
__device__ __forceinline__ void wait_async_lds() {
#if __has_builtin(__builtin_amdgcn_s_wait_asynccnt)
    __builtin_amdgcn_s_wait_asynccnt(0);
#else
    asm volatile("s_wait_asynccnt 0x0" ::: "memory");
#endif
}
#else
#define HAVE_ASYNC_LDS 0
#endif

// ---------------------------------------------------------------------------
// Fused MLP: per block = 16 rows of X; 4 waves each own a 16-wide column tile.
// A-frag (16x4 f32): lane l -> row = l&15, k-pair = (l>>4)*2
// B-frag (4x16 f32): lane l -> col = l&15, k-pair = (l>>4)*2
// C/D (16x16 f32, 8 VGPR): reg r at lane l -> M = (l>>4)*8 + r, N = l&15
// ---------------------------------------------------------------------------
__launch_bounds__(128)
__global__ void mlp_wmma_kernel(const float* __restrict__ X,
                                const float* __restrict__ W1,
                                const float* __restrict__ b1,
                                const float* __restrict__ W2,
                                const float* __restrict__ b2,
                                const float* __restrict__ temp,
                                float* __restrict__ x0,
                                float* __restrict__ hidden) {
    __shared__ float Xs[16 * IN_C];   // 32 KB staged input tile (16B aligned rows)
    __shared__ float Hs[16 * 65];     // padded (stride 65 -> conflict-free)

    const int tid  = threadIdx.x;
    const int wave = tid >> 5;
    const int lane = tid & 31;
    const int m0   = blockIdx.x * 16;

    // stage the 16x512 input tile into LDS
    const float4* Xv = (const float4*)(X + (size_t)m0 * IN_C);
#if HAVE_ASYNC_LDS
    // DMA straight into LDS (global_load_async_to_lds_b128, ASYNCcnt-tracked)
    for (int i = tid; i < 16 * IN_C / 4; i += 128)
        async_copy_b128(&((float4*)Xs)[i], &Xv[i]);
    wait_async_lds();
#else
    float4* Xsv = (float4*)Xs;
    for (int i = tid; i < 16 * IN_C / 4; i += 128) Xsv[i] = Xv[i];
#endif
    __syncthreads();

    const int mA = lane & 15;          // A row
    const int kb = (lane >> 4) * 2;    // which k-pair this half-wave holds
    const int nB = wave * 16 + (lane & 15);  // output column (0..63)

    // ---- Phase 1: H = relu(X @ W1 + b1), one 16x16 tile per wave ----
    v8f acc = {0.f, 0.f, 0.f, 0.f, 0.f, 0.f, 0.f, 0.f};
    for (int kk = 0; kk < IN_C; kk += 4) {
        v2f a, b;
        a.x = Xs[mA * IN_C + kk + kb];
        a.y = Xs[mA * IN_C + kk + kb + 1];
        b.x = W1[(size_t)(kk + kb) * HID + nB];
        b.y = W1[(size_t)(kk + kb + 1) * HID + nB];
        acc = __builtin_amdgcn_wmma_f32_16x16x4_f32(false, a, false, b,
                                                    (short)0, acc, false, false);
    }
    {
        const float bias = b1[nB];
#pragma unroll
        for (int r = 0; r < 8; ++r) {
            const int m = (lane >> 4) * 8 + r;
            const float h = acc[r] + bias;
            Hs[m * 65 + nB] = h > 0.f ? h : 0.f;
        }
    }
    __syncthreads();

    // ---- Phase 2: x0 = H @ W2 + b2 ; hidden = temp[0] * x0 ----
    v8f acc2 = {0.f, 0.f, 0.f, 0.f, 0.f, 0.f, 0.f, 0.f};
    for (int kk = 0; kk < HID; kk += 4) {
        v2f a, b;
        a.x = Hs[mA * 65 + kk + kb];
        a.y = Hs[mA * 65 + kk + kb + 1];
        b.x = W2[(size_t)(kk + kb) * LOC + nB];
        b.y = W2[(size_t)(kk + kb + 1) * LOC + nB];
        acc2 = __builtin_amdgcn_wmma_f32_16x16x4_f32(false, a, false, b,
                                                     (short)0, acc2, false, false);
    }
    {
        const float bias = b2[nB];
        const float t0   = temp[0];
#pragma unroll
        for (int r = 0; r < 8; ++r) {
            const int m = (lane >> 4) * 8 + r;
            const float v = acc2[r] + bias;
            x0[(size_t)(m0 + m) * LOC + nB]     = v;
            hidden[(size_t)(m0 + m) * LOC + nB] = t0 * v;
        }
    }
}

// ---------------------------------------------------------------------------
// GCN normalization: deg initialized to 1 (self loop), count targets, rsqrt.
// ---------------------------------------------------------------------------
__global__ void deg_init_kernel(float* __restrict__ deg, int n) {
    int i = blockIdx.x * blockDim.x + threadIdx.x;
    if (i < n) deg[i] = 1.0f;
}

__global__ void deg_count_kernel(const int* __restrict__ col, float* __restrict__ deg, int E) {
    int e = blockIdx.x * blockDim.x + threadIdx.x;
    if (e < E) atomicAdd(&deg[col[e]], 1.0f);
}

__global__ void deg_rsqrt_kernel(float* __restrict__ deg, int n) {
    int i = blockIdx.x * blockDim.x + threadIdx.x;
    if (i < n) deg[i] = rsqrtf(deg[i]);   // deg >= 1 always (self loop)
}

// ---------------------------------------------------------------------------
// Per-iteration kernels: seed (self-loop term), scatter (edges), axpy (accum).
// ---------------------------------------------------------------------------
__global__ void seed_kernel(const float* __restrict__ src, float* __restrict__ dst,
                            const float* __restrict__ dinv, int n_nodes) {
    int idx = blockIdx.x * blockDim.x + threadIdx.x;   // over n_nodes*16 float4
    if (idx < n_nodes * (LOC / 4)) {
        int node = idx / (LOC / 4);
        float s = dinv[node];
        s *= s;                                        // self-loop norm = dinv^2
        float4 v = ((const float4*)src)[idx];
        v.x *= s; v.y *= s; v.z *= s; v.w *= s;
        ((float4*)dst)[idx] = v;
    }
}

__launch_bounds__(256)
__global__ void scatter_kernel(const int* __restrict__ row, const int* __restrict__ col,
                               const float* __restrict__ dinv,
                               const float* __restrict__ src, float* __restrict__ dst,
                               int E) {
    const int lane = threadIdx.x & 31;
    const long wave = (long)blockIdx.x * (blockDim.x >> 5) + (threadIdx.x >> 5);
    const long e0 = wave * 4;
    // hide edge-list latency across the wave's batch (gfx1250 global_prefetch_b8)
    __builtin_prefetch(row + e0 + 512, 0, 1);
    __builtin_prefetch(col + e0 + 512, 0, 1);
#pragma unroll
    for (int j = 0; j < 4; ++j) {
        const long e = e0 + j;
        if (e < E) {
            const int r = row[e];                     // uniform across wave
            const int c = col[e];
            const float nrm = dinv[r] * dinv[c];
            const float v0 = src[(size_t)r * LOC + lane];        // 128B coalesced
            const float v1 = src[(size_t)r * LOC + 32 + lane];
            atomicAdd(&dst[(size_t)c * LOC + lane],      nrm * v0);
            atomicAdd(&dst[(size_t)c * LOC + 32 + lane], nrm * v1);
        }
    }
}

__global__ void axpy_kernel(float* __restrict__ hidden, const float* __restrict__ buf,
                            const float* __restrict__ temp, int k, int n4) {
    int i = blockIdx.x * blockDim.x + threadIdx.x;    // over n_nodes*16 float4
    if (i < n4) {
        const float t = temp[k];
        float4 h = ((const float4*)hidden)[i];
        float4 v = ((const float4*)buf)[i];
        h.x += t * v.x; h.y += t * v.y; h.z += t * v.z; h.w += t * v.w;
        ((float4*)hidden)[i] = h;
    }
}

// ---------------------------------------------------------------------------
// Final: out = concat(hidden, glob) @ Wl + bl   ([N,128] @ [128,40])
// 3 waves per block cover columns 0..47; cols >= 40 are zero-padded/masked.
// ---------------------------------------------------------------------------
__launch_bounds__(96)
__global__ void final_wmma_kernel(const float* __restrict__ hidden,
                                  const float* __restrict__ glob,
                                  const float* __restrict__ Wl,
                                  const float* __restrict__ bl,
                                  float* __restrict__ out) {
    __shared__ float Es[16 * 129];    // 16 x 128 concat tile, padded stride

    const int tid  = threadIdx.x;
    const int wave = tid >> 5;
    const int lane = tid & 31;
    const int m0   = blockIdx.x * 16;

    for (int i = tid; i < 16 * 32; i += 96) {         // 512 float4 = 16x128
        const int m = i >> 5;
        const int q = i & 31;
        float4 v = (q < 16)
            ? ((const float4*)(hidden + (size_t)(m0 + m) * LOC))[q]
            : ((const float4*)(glob   + (size_t)(m0 + m) * GLOB))[q - 16];
        float* dstp = &Es[m * 129 + q * 4];
        dstp[0] = v.x; dstp[1] = v.y; dstp[2] = v.z; dstp[3] = v.w;
    }
    __syncthreads();

    const int mA = lane & 15;
    const int kb = (lane >> 4) * 2;
    const int n  = wave * 16 + (lane & 15);           // 0..47

    v8f acc = {0.f, 0.f, 0.f, 0.f, 0.f, 0.f, 0.f, 0.f};
    for (int kk = 0; kk < LOC + GLOB; kk += 4) {
        v2f a, b;
        a.x = Es[mA * 129 + kk + kb];
        a.y = Es[mA * 129 + kk + kb + 1];
        b.x = (n < OUT_C) ? Wl[(size_t)(kk + kb) * OUT_C + n] : 0.f;
        b.y = (n < OUT_C) ? Wl[(size_t)(kk + kb + 1) * OUT_C + n] : 0.f;
        acc = __builtin_amdgcn_wmma_f32_16x16x4_f32(false, a, false, b,
                                                    (short)0, acc, false, false);
    }
    if (n < OUT_C) {
        const float bias = bl[n];
#pragma unroll
        for (int r = 0; r < 8; ++r) {
            const int m = (lane >> 4) * 8 + r;
            out[(size_t)(m0 + m) * OUT_C + n] = acc[r] + bias;
        }
    }
}

// ---------------------------------------------------------------------------
extern "C" void kernel_launch(void* const* d_in, const int* in_sizes, int n_in,
                              void* d_out, int out_size, void* d_ws, size_t ws_size,
                              hipStream_t stream) {
    (void)n_in; (void)out_size; (void)ws_size;

    const float* X    = (const float*)d_in[0];
    const int*   ei   = (const int*)  d_in[1];
    const float* glob = (const float*)d_in[2];
    const float* W1   = (const float*)d_in[3];
    const float* b1   = (const float*)d_in[4];
    const float* W2   = (const float*)d_in[5];
    const float* b2   = (const float*)d_in[6];
    const float* temp = (const float*)d_in[7];
    const float* Wl   = (const float*)d_in[8];
    const float* bl   = (const float*)d_in[9];

    const int N = in_sizes[0] / IN_C;     // 100000 (multiple of 16)
    const int E = in_sizes[1] / 2;        // 3200000
    const int* row = ei;                  // sources
    const int* col = ei + E;              // targets

    float* bufA   = (float*)d_ws;
    float* bufB   = bufA + (size_t)N * LOC;
    float* hidden = bufB + (size_t)N * LOC;
    float* dinv   = hidden + (size_t)N * LOC;

    // GCN symmetric normalization
    deg_init_kernel <<<(N + 255) / 256, 256, 0, stream>>>(dinv, N);
    deg_count_kernel<<<(E + 255) / 256, 256, 0, stream>>>(col, dinv, E);
    deg_rsqrt_kernel<<<(N + 255) / 256, 256, 0, stream>>>(dinv, N);

    // Fused MLP (WMMA): bufA = x0, hidden = temp[0] * x0
    mlp_wmma_kernel<<<N / 16, 128, 0, stream>>>(X, W1, b1, W2, b2, temp, bufA, hidden);

    // K rounds of normalized propagation
    const long waves   = ((long)E + 3) / 4;
    const int  sblocks = (int)((waves * 32 + 255) / 256);
    const int  n4      = N * (LOC / 4);
    float* src = bufA;
    float* dst = bufB;
    for (int k = 0; k < KPOW; ++k) {
        seed_kernel   <<<(n4 + 255) / 256, 256, 0, stream>>>(src, dst, dinv, N);
        scatter_kernel<<<sblocks, 256, 0, stream>>>(row, col, dinv, src, dst, E);
        axpy_kernel   <<<(n4 + 255) / 256, 256, 0, stream>>>(hidden, dst, temp, k + 1, n4);
        float* t = src; src = dst; dst = t;
    }

    // Final projection (WMMA)
    final_wmma_kernel<<<N / 16, 96, 0, stream>>>(hidden, glob, Wl, bl, (float*)d_out);
}